// TVLoss_22213570855377
// MI455X (gfx1250) — compile-verified
//
#include <hip/hip_runtime.h>

// Problem constants from the reference: (B,C,H,W) = (32,3,512,512), fp32.
constexpr int Bdim = 32, Cdim = 3, Hdim = 512, Wdim = 512;
constexpr int W4      = Wdim / 4;            // 128 float4 per row
constexpr int PLANE4  = Hdim * W4;           // 65536 float4 per (b,c) plane
constexpr int NPLANES = Bdim * Cdim;         // 96
constexpr int TOTAL4  = NPLANES * PLANE4;    // 6,291,456 float4 total
constexpr int BLOCKS  = 1024;
constexpr int THREADS = 256;                 // 8 wave32s per block

typedef __attribute__((ext_vector_type(2))) float v2f_;
typedef __attribute__((ext_vector_type(8))) float v8f_;

// Full wave32 sum reduction via V_WMMA_F32_16X16X4_F32.
// A(16x4): lane's partial in VGPR0, 0 in VGPR1 -> each partial appears once.
// B(4x16): all ones -> D[m,n] = rowsum_m(A) for every n.
// Lane t8 = sum of its 8 D VGPRs covers half the rows; xor-16 pairs halves.
__device__ __forceinline__ float wave_sum_wmma(float s) {
    v2f_ a; a[0] = s;    a[1] = 0.0f;
    v2f_ b; b[0] = 1.0f; b[1] = 1.0f;
    v8f_ c = {};
    v8f_ d = __builtin_amdgcn_wmma_f32_16x16x4_f32(
        /*neg_a=*/false, a, /*neg_b=*/false, b,
        /*c_mod=*/(short)0, c, /*reuse_a=*/false, /*reuse_b=*/false);
    float t = ((d[0] + d[1]) + (d[2] + d[3])) + ((d[4] + d[5]) + (d[6] + d[7]));
    t += __shfl_xor(t, 16, 32);
    return t;   // every lane now holds the full 32-lane sum
}

__global__ void __launch_bounds__(THREADS)
tv_partial_kernel(const float* __restrict__ x, float* __restrict__ partial) {
    const float4* __restrict__ x4 = reinterpret_cast<const float4*>(x);
    const int tid    = blockIdx.x * THREADS + threadIdx.x;
    const int stride = BLOCKS * THREADS;

    float s = 0.0f;
    // TOTAL4 % stride == 0 -> uniform trip count (24), EXEC stays all-1s.
    for (int g = tid; g < TOTAL4; g += stride) {
        const int plane = g >> 16;              // / PLANE4
        const int rem   = g & (PLANE4 - 1);
        const int row   = rem >> 7;             // / W4
        const int col4  = rem & (W4 - 1);
        const int base4 = plane << 16;

        float4 a  = x4[g];                                                    // self
        float4 dn = x4[base4 + (((row + 1) & (Hdim - 1)) << 7) + col4];       // row below (wrap)
        // column-wrap scalar: element (4*col4 + 4) mod 512 of this row
        const float rx = x[((size_t)(plane * Hdim + row) << 9) + ((4 * col4 + 4) & (Wdim - 1))];

        const float dx0 = a.x - a.y, dx1 = a.y - a.z, dx2 = a.z - a.w, dx3 = a.w - rx;
        const float dy0 = a.x - dn.x, dy1 = a.y - dn.y, dy2 = a.z - dn.z, dy3 = a.w - dn.w;

        s += sqrtf(dx0 * dx0 + dy0 * dy0);
        s += sqrtf(dx1 * dx1 + dy1 * dy1);
        s += sqrtf(dx2 * dx2 + dy2 * dy2);
        s += sqrtf(dx3 * dx3 + dy3 * dy3);
    }

    // Wave32 reduction through the WMMA unit, then cross-wave via LDS.
    const float w = wave_sum_wmma(s);

    __shared__ float lds[THREADS / 32];
    const int lane = threadIdx.x & 31;
    const int wid  = threadIdx.x >> 5;
    if (lane == 0) lds[wid] = w;
    __syncthreads();
    if (threadIdx.x == 0) {
        float t = 0.0f;
        #pragma unroll
        for (int i = 0; i < THREADS / 32; ++i) t += lds[i];
        partial[blockIdx.x] = t;   // every slot overwritten each call: deterministic
    }
}

__global__ void __launch_bounds__(THREADS)
tv_final_kernel(const float* __restrict__ partial, float* __restrict__ out) {
    float s = 0.0f;
    for (int i = threadIdx.x; i < BLOCKS; i += THREADS) s += partial[i];

    #pragma unroll
    for (int off = 16; off > 0; off >>= 1) s += __shfl_down(s, off, 32);

    __shared__ float lds[THREADS / 32];
    const int lane = threadIdx.x & 31;
    const int wid  = threadIdx.x >> 5;
    if (lane == 0) lds[wid] = s;
    __syncthreads();
    if (threadIdx.x == 0) {
        float t = 0.0f;
        #pragma unroll
        for (int i = 0; i < THREADS / 32; ++i) t += lds[i];
        out[0] = t * (1.0f / (float)((long long)Bdim * Cdim * Hdim * Wdim));
    }
}

extern "C" void kernel_launch(void* const* d_in, const int* in_sizes, int n_in,
                              void* d_out, int out_size, void* d_ws, size_t ws_size,
                              hipStream_t stream) {
    const float* x = (const float*)d_in[0];   // (32,3,512,512) fp32, row-major
    float* partial = (float*)d_ws;            // BLOCKS floats of scratch
    float* out     = (float*)d_out;           // single fp32 scalar

    tv_partial_kernel<<<BLOCKS, THREADS, 0, stream>>>(x, partial);
    tv_final_kernel<<<1, THREADS, 0, stream>>>(partial, out);
}